// SetAbstraction_41154376630662
// MI455X (gfx1250) — compile-verified
//
#include <hip/hip_runtime.h>
#include <hip/hip_bf16.h>

// ---------------- problem constants ----------------
constexpr int BB   = 4;
constexpr int NN   = 8192;
constexpr int SS   = 2048;          // N / STRIDE
constexpr int DIN  = 64;
constexpr int MID  = 64;
constexpr int DOUT = 128;
constexpr int NS   = 32;            // nsample
constexpr int K1   = 67;            // D_IN + 3
constexpr float R2   = 0.01f;       // radius^2
constexpr float EPSF = 1e-5f;
constexpr int NROWS = BB * SS * NS; // 262144 rows for MLP
constexpr float INV_NBN = 1.0f / (float)NROWS;

typedef float v2f __attribute__((ext_vector_type(2)));
typedef float v8f __attribute__((ext_vector_type(8)));

static __device__ __forceinline__ v8f wmma4(v2f a, v2f b, v8f c) {
    // V_WMMA_F32_16X16X4_F32 : D = A(16x4) * B(4x16) + C(16x16)
    return __builtin_amdgcn_wmma_f32_16x16x4_f32(
        /*neg_a=*/false, a, /*neg_b=*/false, b,
        /*c_mod=*/(short)0, c, /*reuse_a=*/false, /*reuse_b=*/false);
}

// ---------------- 1) farthest point sampling ----------------
// one block per batch, 1024 threads, 8 points per thread kept in registers
__global__ __launch_bounds__(1024)
void fps_kernel(const float* __restrict__ pos,
                int* __restrict__ fpsIdx,
                float* __restrict__ newxyz /* d_out front */) {
    const int b   = blockIdx.x;
    const int tid = threadIdx.x;
    __shared__ float s_val[32];
    __shared__ int   s_idx[32];
    __shared__ float s_c[3];
    __shared__ int   s_far;

    float px[8], py[8], pz[8], dist[8];
#pragma unroll
    for (int j = 0; j < 8; ++j) {
        int n = tid + j * 1024;
        const float* p = &pos[(size_t)(b * NN + n) * 3];
        px[j] = p[0]; py[j] = p[1]; pz[j] = p[2];
        dist[j] = 1e10f;
    }
    int far = 0;
    for (int i = 0; i < SS; ++i) {
        if (tid == 0) {
            fpsIdx[b * SS + i] = far;
            const float* c = &pos[(size_t)(b * NN + far) * 3];
            float cx = c[0], cy = c[1], cz = c[2];
            float* o = &newxyz[(size_t)(b * SS + i) * 3];
            o[0] = cx; o[1] = cy; o[2] = cz;
            s_c[0] = cx; s_c[1] = cy; s_c[2] = cz;
        }
        __syncthreads();
        const float cx = s_c[0], cy = s_c[1], cz = s_c[2];
        float bv = -1.0f; int bi = 0x7fffffff;
#pragma unroll
        for (int j = 0; j < 8; ++j) {
            float dx = px[j] - cx, dy = py[j] - cy, dz = pz[j] - cz;
            float d = dx * dx + dy * dy + dz * dz;
            dist[j] = fminf(dist[j], d);
            if (dist[j] > bv) { bv = dist[j]; bi = tid + j * 1024; }
        }
        // wave32 reduction (argmax, lowest index on tie)
#pragma unroll
        for (int off = 16; off >= 1; off >>= 1) {
            float ov = __shfl_down(bv, off);
            int   oi = __shfl_down(bi, off);
            if (ov > bv || (ov == bv && oi < bi)) { bv = ov; bi = oi; }
        }
        if ((tid & 31) == 0) { s_val[tid >> 5] = bv; s_idx[tid >> 5] = bi; }
        __syncthreads();
        if (tid < 32) {
            bv = s_val[tid]; bi = s_idx[tid];
#pragma unroll
            for (int off = 16; off >= 1; off >>= 1) {
                float ov = __shfl_down(bv, off);
                int   oi = __shfl_down(bi, off);
                if (ov > bv || (ov == bv && oi < bi)) { bv = ov; bi = oi; }
            }
            if (tid == 0) s_far = bi;
        }
        __syncthreads();
        far = s_far;
    }
}

// ---------------- 2) ball query (one wave per center) ----------------
__global__ __launch_bounds__(256)
void ball_kernel(const float* __restrict__ pos,
                 const float* __restrict__ newxyz,
                 int* __restrict__ ball) {
    const int gw   = (blockIdx.x * blockDim.x + threadIdx.x) >> 5; // query id
    const int lane = threadIdx.x & 31;
    if (gw >= BB * SS) return;
    const int b = gw >> 11;                 // SS == 2048
    const float* c = &newxyz[(size_t)gw * 3];
    const float cx = c[0], cy = c[1], cz = c[2];

    int count = 0, firstIdx = -1;
    for (int c0 = 0; c0 < NN && count < NS; c0 += 32) {
        const int n = c0 + lane;
        const float* p = &pos[(size_t)(b * NN + n) * 3];
        float dx = p[0] - cx, dy = p[1] - cy, dz = p[2] - cz;
        float d = dx * dx + dy * dy + dz * dz;
        unsigned m32 = (unsigned)__ballot(d <= R2);
        if (firstIdx < 0 && m32) firstIdx = c0 + __ffs(m32) - 1;
        int prefix = __popc(m32 & ((1u << lane) - 1u));
        if (((m32 >> lane) & 1u) && (count + prefix) < NS)
            ball[(size_t)gw * NS + count + prefix] = n;
        count += __popc(m32);
    }
    if (count > NS) count = NS;
    if (firstIdx < 0) firstIdx = 0;
    for (int k = count + lane; k < NS; k += 32)
        ball[(size_t)gw * NS + k] = firstIdx;
}

// ---------------- shared GEMM1 tile loader ----------------
static __device__ __forceinline__
void load_group_tile(float* Ash, const float* pos, const float* feat,
                     const float* newxyz, const int* ball,
                     int tid, int block) {
    const int r = tid & 63, half = tid >> 6;
    const int grow = block * 64 + r;       // global MLP row
    const int sg = grow >> 5;              // center id = b*SS+s
    const int b  = sg >> 11;
    const int pidx = ball[grow];
    const float* cc = &newxyz[(size_t)sg * 3];
    const float cx = cc[0], cy = cc[1], cz = cc[2];
    const float* pp = &pos[(size_t)(b * NN + pidx) * 3];
    const float* ff = &feat[(size_t)(b * NN + pidx) * 64];
    __builtin_prefetch(ff, 0, 0);
    float* arow = &Ash[r * 68];
    for (int col = half * 34; col < half * 34 + 34; ++col) {
        float v;
        if (col < 3)       v = pp[col] - ((col == 0) ? cx : (col == 1) ? cy : cz);
        else if (col < K1) v = ff[col - 3];
        else               v = 0.0f;       // K padding (col 67)
        arow[col] = v;
    }
}

// ---------------- 3) GEMM1 statistics pass (WMMA f32) ----------------
__global__ __launch_bounds__(128)
void gemm1_stats(const float* __restrict__ pos, const float* __restrict__ feat,
                 const float* __restrict__ newxyz, const int* __restrict__ ball,
                 const float* __restrict__ W1,
                 float* __restrict__ gsum1, float* __restrict__ gsq1) {
    __shared__ float Ash[64 * 68];
    __shared__ float Bsh[68 * 64];
    __shared__ float s_sum[64], s_sq[64];
    const int tid = threadIdx.x;
    if (tid < 64) { s_sum[tid] = 0.0f; s_sq[tid] = 0.0f; }
    for (int e = tid; e < 68 * 64; e += 128) {
        int k = e >> 6;
        Bsh[e] = (k < K1) ? W1[e] : 0.0f;
    }
    load_group_tile(Ash, pos, feat, newxyz, ball, tid, blockIdx.x);
    __syncthreads();

    const int w = tid >> 5, lane = tid & 31, lo = lane & 15, hi = lane >> 4;
    for (int nt = 0; nt < 4; ++nt) {
        v8f c = {};
        const float* ap = &Ash[(w * 16 + lo) * 68 + 2 * hi];
        const float* bp = &Bsh[(2 * hi) * 64 + nt * 16 + lo];
#pragma unroll
        for (int k0 = 0; k0 < 68; k0 += 4) {
            v2f a;  a.x  = ap[k0];        a.y  = ap[k0 + 1];
            v2f bv; bv.x = bp[k0 * 64];   bv.y = bp[k0 * 64 + 64];
            c = wmma4(a, bv, c);
        }
        float ls = 0.0f, lq = 0.0f;
#pragma unroll
        for (int rr = 0; rr < 8; ++rr) { float v = c[rr]; ls += v; lq += v * v; }
        atomicAdd(&s_sum[nt * 16 + lo], ls);
        atomicAdd(&s_sq[nt * 16 + lo], lq);
    }
    __syncthreads();
    if (tid < 64) { atomicAdd(&gsum1[tid], s_sum[tid]); atomicAdd(&gsq1[tid], s_sq[tid]); }
}

// ---------------- 4) BN finalize ----------------
__global__ void bn_finalize(const float* __restrict__ gsum, const float* __restrict__ gsq,
                            const float* __restrict__ gamma, const float* __restrict__ beta,
                            float* __restrict__ a, float* __restrict__ b, int C) {
    int ch = threadIdx.x;
    if (ch >= C) return;
    float mean = gsum[ch] * INV_NBN;
    float var  = gsq[ch] * INV_NBN - mean * mean;
    float ai   = gamma[ch] * rsqrtf(var + EPSF);
    a[ch] = ai;
    b[ch] = beta[ch] - mean * ai;
}

// ---------------- 5) GEMM1(recompute)+BN1+ReLU -> GEMM2 + stats + pool ----------------
__global__ __launch_bounds__(128)
void gemm2_kernel(const float* __restrict__ pos, const float* __restrict__ feat,
                  const float* __restrict__ newxyz, const int* __restrict__ ball,
                  const float* __restrict__ W1, const float* __restrict__ W2,
                  const float* __restrict__ a1, const float* __restrict__ b1,
                  float* __restrict__ gsum2, float* __restrict__ gsq2,
                  float* __restrict__ pmax, float* __restrict__ pmin) {
    __shared__ float Ash[64 * 68];          // X tile, then h1relu tile
    __shared__ float Bsh[64 * 132];         // W1 (68x64) then W2 (64x132)
    __shared__ float s_sum[128], s_sq[128];
    __shared__ float s_max[4 * 128], s_min[4 * 128];
    const int tid = threadIdx.x;
    if (tid < 128) { s_sum[tid] = 0.0f; s_sq[tid] = 0.0f; }
    for (int e = tid; e < 68 * 64; e += 128) {
        int k = e >> 6;
        Bsh[e] = (k < K1) ? W1[e] : 0.0f;
    }
    load_group_tile(Ash, pos, feat, newxyz, ball, tid, blockIdx.x);
    __syncthreads();

    const int w = tid >> 5, lane = tid & 31, lo = lane & 15, hi = lane >> 4;
    v8f acc[4];
    for (int nt = 0; nt < 4; ++nt) {
        v8f c = {};
        const float* ap = &Ash[(w * 16 + lo) * 68 + 2 * hi];
        const float* bp = &Bsh[(2 * hi) * 64 + nt * 16 + lo];
#pragma unroll
        for (int k0 = 0; k0 < 68; k0 += 4) {
            v2f a;  a.x  = ap[k0];      a.y  = ap[k0 + 1];
            v2f bv; bv.x = bp[k0 * 64]; bv.y = bp[k0 * 64 + 64];
            c = wmma4(a, bv, c);
        }
        acc[nt] = c;
    }
    __syncthreads();  // done reading X tile and W1

    // write h1relu in A layout; load W2
    for (int nt = 0; nt < 4; ++nt) {
        const int ch = nt * 16 + lo;
        const float ga = a1[ch], gb = b1[ch];
#pragma unroll
        for (int rr = 0; rr < 8; ++rr) {
            float v = fmaxf(0.0f, ga * acc[nt][rr] + gb);
            Ash[(w * 16 + rr + 8 * hi) * 68 + ch] = v;
        }
    }
    for (int e = tid; e < 64 * 132; e += 128) {
        int k = e / 132, n = e % 132;
        Bsh[e] = (n < DOUT) ? W2[k * DOUT + n] : 0.0f;
    }
    __syncthreads();

    // GEMM2: (64 x 64) @ (64 x 128)
    for (int nt2 = 0; nt2 < 8; ++nt2) {
        v8f c = {};
        const float* ap = &Ash[(w * 16 + lo) * 68 + 2 * hi];
        const float* bp = &Bsh[(2 * hi) * 132 + nt2 * 16 + lo];
#pragma unroll
        for (int k0 = 0; k0 < 64; k0 += 4) {
            v2f a;  a.x  = ap[k0];         a.y  = ap[k0 + 1];
            v2f bv; bv.x = bp[k0 * 132];   bv.y = bp[k0 * 132 + 132];
            c = wmma4(a, bv, c);
        }
        const int ch = nt2 * 16 + lo;
        float ls = 0.0f, lq = 0.0f;
        float mx = -1e30f, mn = 1e30f;
#pragma unroll
        for (int rr = 0; rr < 8; ++rr) {
            float v = c[rr];
            ls += v; lq += v * v;
            mx = fmaxf(mx, v); mn = fminf(mn, v);
        }
        atomicAdd(&s_sum[ch], ls);
        atomicAdd(&s_sq[ch], lq);
        mx = fmaxf(mx, __shfl_xor(mx, 16));
        mn = fminf(mn, __shfl_xor(mn, 16));
        if (hi == 0) { s_max[w * 128 + ch] = mx; s_min[w * 128 + ch] = mn; }
    }
    __syncthreads();

    // combine the two M-tiles of each center, write pooled max/min
    for (int e = tid; e < 256; e += 128) {
        int c_ = e >> 7, ch = e & 127;
        float mx = fmaxf(s_max[(2 * c_) * 128 + ch], s_max[(2 * c_ + 1) * 128 + ch]);
        float mn = fminf(s_min[(2 * c_) * 128 + ch], s_min[(2 * c_ + 1) * 128 + ch]);
        size_t cidx = (size_t)blockIdx.x * 2 + c_;
        pmax[cidx * DOUT + ch] = mx;
        pmin[cidx * DOUT + ch] = mn;
    }
    if (tid < 128) { atomicAdd(&gsum2[tid], s_sum[tid]); atomicAdd(&gsq2[tid], s_sq[tid]); }
}

// ---------------- 6) skip GEMM: feature[fps_idx] @ Wskip + bskip ----------------
__global__ __launch_bounds__(128)
void skip_gemm(const float* __restrict__ feat, const int* __restrict__ fpsIdx,
               const float* __restrict__ Wsk, const float* __restrict__ bsk,
               float* __restrict__ skipb) {
    __shared__ float Ash[64 * 68];
    __shared__ float Bsh[64 * 132];
    const int tid = threadIdx.x;
    for (int e = tid; e < 64 * 132; e += 128) {
        int k = e / 132, n = e % 132;
        Bsh[e] = (n < DOUT) ? Wsk[k * DOUT + n] : 0.0f;
    }
    const int r = tid & 63, half = tid >> 6;
    const int cidx = blockIdx.x * 64 + r;
    const int b = cidx >> 11;
    const int pidx = fpsIdx[cidx];
    const float* ff = &feat[(size_t)(b * NN + pidx) * 64];
    for (int col = half * 32; col < half * 32 + 32; ++col)
        Ash[r * 68 + col] = ff[col];
    __syncthreads();

    const int w = tid >> 5, lane = tid & 31, lo = lane & 15, hi = lane >> 4;
    for (int nt = 0; nt < 8; ++nt) {
        v8f c = {};
        const float* ap = &Ash[(w * 16 + lo) * 68 + 2 * hi];
        const float* bp = &Bsh[(2 * hi) * 132 + nt * 16 + lo];
#pragma unroll
        for (int k0 = 0; k0 < 64; k0 += 4) {
            v2f a;  a.x  = ap[k0];       a.y  = ap[k0 + 1];
            v2f bv; bv.x = bp[k0 * 132]; bv.y = bp[k0 * 132 + 132];
            c = wmma4(a, bv, c);
        }
        const int ch = nt * 16 + lo;
        const float bias = bsk[ch];
#pragma unroll
        for (int rr = 0; rr < 8; ++rr) {
            size_t row = (size_t)blockIdx.x * 64 + w * 16 + rr + 8 * hi;
            skipb[row * DOUT + ch] = c[rr] + bias;
        }
    }
}

// ---------------- 7) final: bn2 on pooled value + skip + relu ----------------
__global__ __launch_bounds__(256)
void final_kernel(const float* __restrict__ pmax, const float* __restrict__ pmin,
                  const float* __restrict__ skipb,
                  const float* __restrict__ a2, const float* __restrict__ b2,
                  float* __restrict__ out) {
    const int t = blockIdx.x * blockDim.x + threadIdx.x;
    if (t >= BB * SS * DOUT) return;
    const int ch = t & (DOUT - 1);
    const float a = a2[ch], b = b2[ch];
    float v = (a >= 0.0f) ? (a * pmax[t] + b) : (a * pmin[t] + b);
    out[t] = fmaxf(0.0f, v + skipb[t]);
}

// ---------------- launcher ----------------
extern "C" void kernel_launch(void* const* d_in, const int* in_sizes, int n_in,
                              void* d_out, int out_size, void* d_ws, size_t ws_size,
                              hipStream_t stream) {
    const float* pos    = (const float*)d_in[0];
    const float* feat   = (const float*)d_in[1];
    const float* W1     = (const float*)d_in[2];
    const float* gamma1 = (const float*)d_in[3];
    const float* beta1  = (const float*)d_in[4];
    const float* W2     = (const float*)d_in[5];
    const float* gamma2 = (const float*)d_in[6];
    const float* beta2  = (const float*)d_in[7];
    const float* Wsk    = (const float*)d_in[8];
    const float* bsk    = (const float*)d_in[9];

    float* newxyz  = (float*)d_out;                 // (B,S,3)
    float* outfeat = (float*)d_out + BB * SS * 3;   // (B,S,128)

    char* w = (char*)d_ws;
    int*   fpsIdx = (int*)w;        w += (size_t)BB * SS * sizeof(int);
    int*   ball   = (int*)w;        w += (size_t)BB * SS * NS * sizeof(int);
    float* gsum1  = (float*)w;      w += 64 * sizeof(float);
    float* gsq1   = (float*)w;      w += 64 * sizeof(float);
    float* gsum2  = (float*)w;      w += 128 * sizeof(float);
    float* gsq2   = (float*)w;      w += 128 * sizeof(float);
    float* a1     = (float*)w;      w += 64 * sizeof(float);
    float* b1     = (float*)w;      w += 64 * sizeof(float);
    float* a2     = (float*)w;      w += 128 * sizeof(float);
    float* b2     = (float*)w;      w += 128 * sizeof(float);
    float* pmax   = (float*)w;      w += (size_t)BB * SS * DOUT * sizeof(float);
    float* pmin   = (float*)w;      w += (size_t)BB * SS * DOUT * sizeof(float);
    float* skipb  = (float*)w;      w += (size_t)BB * SS * DOUT * sizeof(float);

    // zero the stat accumulators (contiguous 384 floats starting at gsum1)
    hipMemsetAsync(gsum1, 0, 384 * sizeof(float), stream);

    fps_kernel<<<BB, 1024, 0, stream>>>(pos, fpsIdx, newxyz);
    ball_kernel<<<(BB * SS * 32) / 256, 256, 0, stream>>>(pos, newxyz, ball);
    gemm1_stats<<<NROWS / 64, 128, 0, stream>>>(pos, feat, newxyz, ball, W1, gsum1, gsq1);
    bn_finalize<<<1, 64, 0, stream>>>(gsum1, gsq1, gamma1, beta1, a1, b1, 64);
    gemm2_kernel<<<NROWS / 64, 128, 0, stream>>>(pos, feat, newxyz, ball, W1, W2,
                                                 a1, b1, gsum2, gsq2, pmax, pmin);
    bn_finalize<<<1, 128, 0, stream>>>(gsum2, gsq2, gamma2, beta2, a2, b2, 128);
    skip_gemm<<<(BB * SS) / 64, 128, 0, stream>>>(feat, fpsIdx, Wsk, bsk, skipb);
    final_kernel<<<(BB * SS * DOUT) / 256, 256, 0, stream>>>(pmax, pmin, skipb, a2, b2, outfeat);
}